// Word2VecCBOW_49589692399746
// MI455X (gfx1250) — compile-verified
//
#include <hip/hip_runtime.h>
#include <hip/hip_bf16.h>

// Problem constants (match reference)
#define BB 64
#define CC 8
#define VV 50000
#define HH 128

typedef __attribute__((ext_vector_type(2))) float v2f;
typedef __attribute__((ext_vector_type(8))) float v8f;

// ---------------------------------------------------------------------------
// K1: S[v] = sum_h W_ctx[v,h].  One wave32 per row; lane loads float4 (32*4=128).
// W_ctx read once: 25.6 MB.
// ---------------------------------------------------------------------------
__global__ __launch_bounds__(256) void rowsum_kernel(
    const float* __restrict__ W_ctx, float* __restrict__ S) {
  const int wave = (blockIdx.x * blockDim.x + threadIdx.x) >> 5;
  const int lane = threadIdx.x & 31;
  if (wave >= VV) return;
  const float4 w = *(const float4*)(W_ctx + (size_t)wave * HH + lane * 4);
  float s = w.x + w.y + w.z + w.w;
#pragma unroll
  for (int off = 16; off > 0; off >>= 1) s += __shfl_down(s, off, 32);
  if (lane == 0) S[wave] = s;
}

// ---------------------------------------------------------------------------
// K2: latent[r] = dot(x[r, 0:V], S),  r = b*C + c  (512 rows of 50000).
// One 256-thread block per row; float4 grid-stride; LDS tree reduce.
// x read once: 102.4 MB (the dominant traffic). S (200 KB) stays hot in L2.
// ---------------------------------------------------------------------------
__global__ __launch_bounds__(256) void latent_kernel(
    const float* __restrict__ x, const float* __restrict__ S,
    float* __restrict__ latent) {
  const int row = blockIdx.x;  // 0..511
  const float4* __restrict__ xr = (const float4*)(x + (size_t)row * VV);
  const float4* __restrict__ sv = (const float4*)S;
  float acc = 0.f;
  for (int i = threadIdx.x; i < VV / 4; i += 256) {
    const float4 a = xr[i];
    const float4 b = sv[i];
    acc += a.x * b.x + a.y * b.y + a.z * b.z + a.w * b.w;
  }
  __shared__ float red[256];
  red[threadIdx.x] = acc;
  __syncthreads();
#pragma unroll
  for (int s = 128; s > 0; s >>= 1) {
    if (threadIdx.x < (unsigned)s) red[threadIdx.x] += red[threadIdx.x + s];
    __syncthreads();
  }
  if (threadIdx.x == 0) latent[row] = red[0];
}

// ---------------------------------------------------------------------------
// K3: logits[m, n] = sum_k latent[m,k] * W_out[n,k] + b_out[n]
//     M=B=64, N=V=50000, K=C=8.
// One wave per 16x16 tile via V_WMMA_F32_16X16X4_F32, K=8 as two k=4 steps.
//
// A (16x4 f32, M x K) per ISA layout: lanes 0-15 -> M=lane, VGPR0/1 = K 0/1;
// lanes 16-31 -> M=lane-16, VGPR0/1 = K 2/3.
// B (4x16 f32, K x N) mirrored: lane indexes N, half-wave selects K pair.
// W_out is [V, C] row-major, so B[k][n] = W_out[(n0+n)*C + k] -> contiguous
// per-lane pair loads.
// C/D (16x16 f32): VGPR r, lanes 0-15 -> (M=r, N=lane); lanes 16-31 ->
// (M=r+8, N=lane-16).
// ---------------------------------------------------------------------------
__global__ __launch_bounds__(256) void logits_wmma_kernel(
    const float* __restrict__ latent, const float* __restrict__ W_out,
    const float* __restrict__ b_out, float* __restrict__ out) {
  const int lane = threadIdx.x & 31;
  const int waveInBlk = threadIdx.x >> 5;
  const int t = blockIdx.x * 8 + waveInBlk;          // tile id
  const int NT = VV / 16;                            // 3125 column tiles
  const int TOT = (BB / 16) * NT;                    // 12500 tiles
  if (t >= TOT) return;                              // wave-uniform: EXEC stays all-1s

  const int mt = t / NT;
  const int nt = t - mt * NT;
  const int m0 = mt * 16;
  const int n0 = nt * 16;

  const int l15 = lane & 15;
  const int kb = (lane >> 4) * 2;  // 0 for lanes 0-15, 2 for lanes 16-31

  // A tiles (two K-halves of latent rows)
  const float* __restrict__ arow = latent + (size_t)(m0 + l15) * CC;
  v2f a0, a1;
  a0.x = arow[kb];     a0.y = arow[kb + 1];
  a1.x = arow[kb + 4]; a1.y = arow[kb + 5];

  // B tiles (W_out rows are the K-vectors for each output column)
  const float* __restrict__ brow = W_out + (size_t)(n0 + l15) * CC;
  v2f b0, b1;
  b0.x = brow[kb];     b0.y = brow[kb + 1];
  b1.x = brow[kb + 4]; b1.y = brow[kb + 5];

  v8f d = {};
  d = __builtin_amdgcn_wmma_f32_16x16x4_f32(false, a0, false, b0,
                                            (short)0, d, false, false);
  d = __builtin_amdgcn_wmma_f32_16x16x4_f32(false, a1, false, b1,
                                            (short)0, d, false, false);

  const float bias = b_out[n0 + l15];
  const int rbase = m0 + (lane >> 4) * 8;
  float* __restrict__ o = out + (size_t)rbase * VV + n0 + l15;
#pragma unroll
  for (int r = 0; r < 8; ++r) {
    o[(size_t)r * VV] = d[r] + bias;
  }
}

// ---------------------------------------------------------------------------
// Launch: inputs in setup_inputs() order: x [B*C*V], W_ctx [V*H],
// W_out [V*C], b_out [V].  Output: logits [B*V] f32.
// Workspace: S (V floats) at d_ws+0, latent (B*C floats) at d_ws+200000
// (16B-aligned since 200000 % 16 == 0).
// ---------------------------------------------------------------------------
extern "C" void kernel_launch(void* const* d_in, const int* in_sizes, int n_in,
                              void* d_out, int out_size, void* d_ws,
                              size_t ws_size, hipStream_t stream) {
  const float* x     = (const float*)d_in[0];
  const float* W_ctx = (const float*)d_in[1];
  const float* W_out = (const float*)d_in[2];
  const float* b_out = (const float*)d_in[3];
  float* logits = (float*)d_out;

  float* S      = (float*)d_ws;                         // 50000 floats
  float* latent = (float*)((char*)d_ws + (size_t)VV * sizeof(float));  // 512 floats

  // K1: 50000 waves, 8 waves/block
  rowsum_kernel<<<(VV + 7) / 8, 256, 0, stream>>>(W_ctx, S);
  // K2: one block per (b,c) row
  latent_kernel<<<BB * CC, 256, 0, stream>>>(x, S, latent);
  // K3: 12500 tiles, 8 waves/block
  const int TOT = (BB / 16) * (VV / 16);
  logits_wmma_kernel<<<(TOT + 7) / 8, 256, 0, stream>>>(latent, W_out, b_out,
                                                        logits);
}